// GlobalGraphNet_8400956031363
// MI455X (gfx1250) — compile-verified
//
#include <hip/hip_runtime.h>
#include <hip/hip_bf16.h>

typedef __attribute__((ext_vector_type(2))) float v2f;
typedef __attribute__((ext_vector_type(8))) float v8f;

#define NNODES 38333
#define NPAD   38336            // multiple of 64 (4 waves x 16 rows / block)
#define NEDGE  (38333 * 32)
#define EMBD   62
#define SLOPE  0.01f

__device__ __forceinline__ float lrelu(float z) { return z >= 0.f ? z : SLOPE * z; }

// ---------------------------------------------------------------- utility
__global__ void k_fill(float* __restrict__ p, float val, int n) {
    int i = blockIdx.x * blockDim.x + threadIdx.x;
    if (i < n) p[i] = val;
}

// ---------------------------------------------------------------- degrees
__global__ void k_deg_accum(const int* __restrict__ ei, float* __restrict__ deg) {
    int e = blockIdx.x * blockDim.x + threadIdx.x;
    if (e < NEDGE) {
        int col = ei[NEDGE + e];
        atomicAdd(&deg[col], 1.0f);
    }
}

__global__ void k_deg_finish(float* __restrict__ dinv) {
    int i = blockIdx.x * blockDim.x + threadIdx.x;
    if (i < NPAD) dinv[i] = rsqrtf(dinv[i]);
}

// ---------------------------------------------------------------- h0 build: [NPAD x 128], col 127 zero pad, rows >= N zero
__global__ void k_embed(const float* __restrict__ x, const float* __restrict__ emb,
                        float* __restrict__ h0) {
    int t = blockIdx.x * blockDim.x + threadIdx.x;
    if (t >= NPAD * 128) return;
    int i = t >> 7, c = t & 127;
    float v = 0.f;
    if (i < NNODES) {
        if (c < EMBD) {
            int id0 = (int)x[i * 5 + 0];
            v = emb[id0 * EMBD + c];
        } else if (c < 2 * EMBD) {
            int id1 = (int)x[i * 5 + 1];
            v = emb[id1 * EMBD + (c - EMBD)];
        } else if (c < 127) {
            v = x[i * 5 + 2 + (c - 2 * EMBD)];
        }
    }
    h0[t] = v;
}

// ---------------------------------------------------------------- W1 pad: [127,64] -> [128,64] zero last row
__global__ void k_padW1(const float* __restrict__ W1, float* __restrict__ Wp) {
    int t = blockIdx.x * blockDim.x + threadIdx.x;
    if (t >= 128 * 64) return;
    int k = t >> 6;
    Wp[t] = (k < 127) ? W1[t] : 0.f;
}

// ---------------------------------------------------------------- WMMA f32 GEMM: HW[NPAD x OUT] = H[NPAD x IN(stride)] @ W[IN x OUT]
// One 16x16 output tile per wave; V_WMMA_F32_16X16X4_F32, K unrolled.
// No bounds checks anywhere -> EXEC all-1s at every WMMA (ISA requirement).
// A frag (16x4): lanes 0-15 hold K=0,1; lanes 16-31 hold K=2,3 (per row M=lane&15)
// B frag (4x16): VGPR0 = rows K0/K2, VGPR1 = rows K1/K3 (N = lane&15)
// D (16x16 f32): acc[v] -> row tileM + v + (lane>=16 ? 8 : 0), col tileN + (lane&15)
template <int IN, int OUT, int STRIDE>
__global__ void k_gemm_wmma(const float* __restrict__ H, const float* __restrict__ W,
                            float* __restrict__ HW) {
    const int lane = threadIdx.x & 31;
    const int wave = threadIdx.x >> 5;
    const int tm = (blockIdx.x * 4 + wave) * 16;
    const int tn = blockIdx.y * 16;
    const int l15 = lane & 15;
    const int hi = lane >> 4;        // 0 or 1
    const int kh = hi * 2;           // K offset within group of 4
    const int r = tm + l15;
    const int c = tn + l15;
    v8f acc = {0.f, 0.f, 0.f, 0.f, 0.f, 0.f, 0.f, 0.f};
#pragma unroll
    for (int kk = 0; kk < IN; kk += 4) {
        v2f a, b;
        a.x = H[(size_t)r * STRIDE + kk + kh];
        a.y = H[(size_t)r * STRIDE + kk + kh + 1];
        b.x = W[(size_t)(kk + kh) * OUT + c];
        b.y = W[(size_t)(kk + kh + 1) * OUT + c];
        acc = __builtin_amdgcn_wmma_f32_16x16x4_f32(false, a, false, b, (short)0, acc,
                                                    false, false);
    }
    const int rb = tm + hi * 8;
#pragma unroll
    for (int v = 0; v < 8; ++v)
        HW[(size_t)(rb + v) * OUT + c] = acc[v];
}

// ---------------------------------------------------------------- edge scatter: agg[col,f] += hw[row,f] * dinv[row]*dinv[col]
// one thread per (edge, feature): coalesced in f, one L2 atomic each
template <int LOG2OUT>
__global__ void k_scatter(const int* __restrict__ ei, const float* __restrict__ dinv,
                          const float* __restrict__ hw, float* __restrict__ agg) {
    const long long total = (long long)NEDGE << LOG2OUT;
    long long t = (long long)blockIdx.x * blockDim.x + threadIdx.x;
    if (t >= total) return;
    int f = (int)(t & ((1 << LOG2OUT) - 1));
    int e = (int)(t >> LOG2OUT);
    int row = ei[e];
    int col = ei[NEDGE + e];
    float nrm = dinv[row] * dinv[col];
    atomicAdd(&agg[((size_t)col << LOG2OUT) + f],
              hw[((size_t)row << LOG2OUT) + f] * nrm);
}

// ---------------------------------------------------------------- epilogue: out = act(agg + hw*dinv^2 + b)
// ACT 0: lrelu(z)   ACT 1: lrelu(z) + z
template <int LOG2OUT, int ACT>
__global__ void k_epilogue(const float* __restrict__ agg, const float* __restrict__ hw,
                           const float* __restrict__ dinv, const float* __restrict__ b,
                           float* __restrict__ out) {
    int t = blockIdx.x * blockDim.x + threadIdx.x;
    if (t >= (NPAD << LOG2OUT)) return;
    int i = t >> LOG2OUT;
    int f = t & ((1 << LOG2OUT) - 1);
    float di = dinv[i];
    float z = agg[t] + hw[t] * di * di + b[f];
    out[t] = (ACT == 0) ? lrelu(z) : (lrelu(z) + z);
}

// ---------------------------------------------------------------- layer-5 gemv: pre[i] = h[i,:32] . W5
__global__ void k_gemv32(const float* __restrict__ H, const float* __restrict__ W5,
                         float* __restrict__ pre) {
    int i = blockIdx.x * blockDim.x + threadIdx.x;
    if (i >= NPAD) return;
    float s = 0.f;
#pragma unroll
    for (int k = 0; k < 32; ++k) s += H[(size_t)i * 32 + k] * W5[k];
    pre[i] = s;
}

// ---------------------------------------------------------------- fc1: o[j] = sum_i v[i]*Wf1[i,j]  (block-partial + atomic)
__global__ void k_fc1(const float* __restrict__ v, const float* __restrict__ Wf1,
                      float* __restrict__ o) {
    int j = threadIdx.x;  // 0..127
    int rpb = (NNODES + gridDim.x - 1) / gridDim.x;
    int i0 = blockIdx.x * rpb;
    int i1 = min(NNODES, i0 + rpb);
    float acc = 0.f;
    for (int i = i0; i < i1; ++i) acc += v[i] * Wf1[(size_t)i * 128 + j];
    atomicAdd(&o[j], acc);
}

// ---------------------------------------------------------------- fc2: out = relu(relu(o1+bf1) @ Wf2 + bf2), single block of 128
__global__ void k_fc2(const float* __restrict__ o1, const float* __restrict__ bf1,
                      const float* __restrict__ Wf2, const float* __restrict__ bf2,
                      float* __restrict__ out) {
    __shared__ float a[128];
    int j = threadIdx.x;
    a[j] = fmaxf(0.f, o1[j] + bf1[j]);
    __syncthreads();
    float acc = 0.f;
#pragma unroll 8
    for (int k = 0; k < 128; ++k) acc += a[k] * Wf2[(size_t)k * 128 + j];
    out[j] = fmaxf(0.f, acc + bf2[j]);
}

// ================================================================ launch
static inline int cdiv(long long a, int b) { return (int)((a + b - 1) / b); }

extern "C" void kernel_launch(void* const* d_in, const int* in_sizes, int n_in,
                              void* d_out, int out_size, void* d_ws, size_t ws_size,
                              hipStream_t stream) {
    const float* x    = (const float*)d_in[0];
    const int*   ei   = (const int*)d_in[1];
    const float* emb  = (const float*)d_in[2];
    const float* W1   = (const float*)d_in[3];
    const float* b1   = (const float*)d_in[4];
    const float* W2   = (const float*)d_in[5];
    const float* b2   = (const float*)d_in[6];
    const float* W3   = (const float*)d_in[7];
    const float* b3   = (const float*)d_in[8];
    const float* W4   = (const float*)d_in[9];
    const float* b4   = (const float*)d_in[10];
    const float* W5   = (const float*)d_in[11];
    const float* b5   = (const float*)d_in[12];
    const float* Wf1  = (const float*)d_in[13];
    const float* bf1  = (const float*)d_in[14];
    const float* Wf2  = (const float*)d_in[15];
    const float* bf2  = (const float*)d_in[16];
    float* out = (float*)d_out;

    // workspace layout (floats)
    float* ws   = (float*)d_ws;
    float* dinv = ws;                        // NPAD
    float* hA   = dinv + NPAD;               // NPAD*128
    float* hB   = hA + (size_t)NPAD * 128;   // NPAD*64  (hw buffer)
    float* hC   = hB + (size_t)NPAD * 64;    // NPAD*64  (agg buffer)
    float* hD   = hC + (size_t)NPAD * 64;    // NPAD*64  (ping-pong h)
    float* Wp   = hD + (size_t)NPAD * 64;    // 128*64 padded W1
    float* obuf = Wp + 128 * 64;             // 128

    const int T = 256;

    // --- degrees: deg = 1 (self loop) + scatter, then rsqrt in place
    k_fill<<<cdiv(NPAD, T), T, 0, stream>>>(dinv, 1.0f, NPAD);
    k_deg_accum<<<cdiv(NEDGE, T), T, 0, stream>>>(ei, dinv);
    k_deg_finish<<<cdiv(NPAD, T), T, 0, stream>>>(dinv);

    // --- node features h0 [NPAD x 128] (col 127 zero), padded W1
    k_embed<<<cdiv((long long)NPAD * 128, T), T, 0, stream>>>(x, emb, hA);
    k_padW1<<<cdiv(128 * 64, T), T, 0, stream>>>(W1, Wp);

    dim3 gb(128);  // 4 waves per block
    const int MB = NPAD / 64;  // 599 blocks in M

    // --- layer 1: IN 128(pad), OUT 64, h: hA(stride128) -> hD
    k_gemm_wmma<128, 64, 128><<<dim3(MB, 4), gb, 0, stream>>>(hA, Wp, hB);
    k_fill<<<cdiv((long long)NPAD * 64, T), T, 0, stream>>>(hC, 0.f, NPAD * 64);
    k_scatter<6><<<cdiv((long long)NEDGE * 64, T), T, 0, stream>>>(ei, dinv, hB, hC);
    k_epilogue<6, 0><<<cdiv((long long)NPAD * 64, T), T, 0, stream>>>(hC, hB, dinv, b1, hD);

    // --- layer 2: IN 64, OUT 32, hD -> hA
    k_gemm_wmma<64, 32, 64><<<dim3(MB, 2), gb, 0, stream>>>(hD, W2, hB);
    k_fill<<<cdiv((long long)NPAD * 32, T), T, 0, stream>>>(hC, 0.f, NPAD * 32);
    k_scatter<5><<<cdiv((long long)NEDGE * 32, T), T, 0, stream>>>(ei, dinv, hB, hC);
    k_epilogue<5, 0><<<cdiv((long long)NPAD * 32, T), T, 0, stream>>>(hC, hB, dinv, b2, hA);

    // --- layer 3: IN 32, OUT 32, residual act, hA -> hD
    k_gemm_wmma<32, 32, 32><<<dim3(MB, 2), gb, 0, stream>>>(hA, W3, hB);
    k_fill<<<cdiv((long long)NPAD * 32, T), T, 0, stream>>>(hC, 0.f, NPAD * 32);
    k_scatter<5><<<cdiv((long long)NEDGE * 32, T), T, 0, stream>>>(ei, dinv, hB, hC);
    k_epilogue<5, 1><<<cdiv((long long)NPAD * 32, T), T, 0, stream>>>(hC, hB, dinv, b3, hD);

    // --- layer 4: IN 32, OUT 32, residual act, hD -> hA
    k_gemm_wmma<32, 32, 32><<<dim3(MB, 2), gb, 0, stream>>>(hD, W4, hB);
    k_fill<<<cdiv((long long)NPAD * 32, T), T, 0, stream>>>(hC, 0.f, NPAD * 32);
    k_scatter<5><<<cdiv((long long)NEDGE * 32, T), T, 0, stream>>>(ei, dinv, hB, hC);
    k_epilogue<5, 1><<<cdiv((long long)NPAD * 32, T), T, 0, stream>>>(hC, hB, dinv, b4, hA);

    // --- layer 5: IN 32, OUT 1, hA -> v in hD
    k_gemv32<<<cdiv(NPAD, T), T, 0, stream>>>(hA, W5, hB);
    k_fill<<<cdiv(NPAD, T), T, 0, stream>>>(hC, 0.f, NPAD);
    k_scatter<0><<<cdiv(NEDGE, T), T, 0, stream>>>(ei, dinv, hB, hC);
    k_epilogue<0, 0><<<cdiv(NPAD, T), T, 0, stream>>>(hC, hB, dinv, b5, hD);

    // --- fc head
    k_fill<<<1, T, 0, stream>>>(obuf, 0.f, 128);
    k_fc1<<<256, 128, 0, stream>>>(hD, Wf1, obuf);
    k_fc2<<<1, 128, 0, stream>>>(obuf, bf1, Wf2, bf2, out);
}